// ARNOLD_ENC_52639119180424
// MI455X (gfx1250) — compile-verified
//
#include <hip/hip_runtime.h>
#include <stdint.h>

// ---------------------------------------------------------------------------
// Spike latency encoder for MI455X (gfx1250).
//
// Output [100, 1024, 1024] f32 = 400 MiB, >=99% zeros -> store-BW bound
// (~17us @ 23.3 TB/s). Strategy:
//   Kernel 1: zero-fill via Tensor Data Mover (TENSOR_STORE_FROM_LDS) streaming
//             a 64KB zeroed LDS tile, 4 iterations per op = 256KB per block.
//   Kernel 2: 1M threads compute bins and store the one-hot 1.0f values
//             (unique (bin,b,n) per (b,n) -> plain stores, no atomics).
// ---------------------------------------------------------------------------

typedef unsigned int v4u __attribute__((ext_vector_type(4)));
typedef int          v8i __attribute__((ext_vector_type(8)));
typedef int          v4i __attribute__((ext_vector_type(4)));

#define TILE_DWORDS 16384                      // 64 KB LDS zero tile
#define TDM_ITER    4                          // iterations per TDM op
#define BLOCK_BYTES (TILE_DWORDS * 4 * TDM_ITER)  // 256 KB per block

__global__ void ARNOLD_zero_fill_tdm(float* __restrict__ out) {
  __shared__ unsigned int lds_zero[TILE_DWORDS];

  // Zero the LDS tile with wide DS stores.
  uint4 z = make_uint4(0u, 0u, 0u, 0u);
  uint4* p = (uint4*)lds_zero;
  const int nvec = TILE_DWORDS / 4;
  for (int i = threadIdx.x; i < nvec; i += blockDim.x) p[i] = z;

  // Escape the LDS buffer so the zeroing stores cannot be dead-store
  // eliminated (the TDM builtin only sees an integer LDS offset, so alias
  // analysis would otherwise delete the DS stores).
  asm volatile("" : : "r"(lds_zero) : "memory");
  __syncthreads();

  if (threadIdx.x == 0) {
    uint64_t gaddr   = (uint64_t)(uintptr_t)out + (uint64_t)blockIdx.x * BLOCK_BYTES;
    // For LDS objects the low 32 bits of the flat address are the logical LDS
    // byte offset (ISA 10.2).
    uint32_t lds_off = (uint32_t)(uintptr_t)lds_zero;

    // ---- D# group 0 (128b): count=1, lds_addr, 57-bit global addr, type=2
    v4u g0;
    g0[0] = 1u;                                        // count = 1, user mode
    g0[1] = lds_off;                                   // lds_addr (bytes)
    g0[2] = (uint32_t)gaddr;                           // global_addr[31:0]
    g0[3] = ((uint32_t)(gaddr >> 32) & 0x01FFFFFFu)    // global_addr[56:32]
            | (2u << 30);                              // type = 2 ("image")

    // ---- D# group 1 (256b)
    v8i g1;
    g1[0] = (int)((2u << 16) | (1u << 19));            // data_size=4B, iterate_enable=1
    g1[1] = (int)((TILE_DWORDS & 0xFFFFu) << 16);      // atomic_bar=0; tensor_dim0 lo16
    g1[2] = (int)((TILE_DWORDS >> 16) | (1u << 16));   // tensor_dim0 hi16; tensor_dim1=1
    g1[3] = (int)((TILE_DWORDS & 0xFFFFu) << 16);      // tensor_dim1 hi16=0; tile_dim0=16384
    g1[4] = 1;                                         // tile_dim1=1; tile_dim2=0
    g1[5] = TILE_DWORDS;                               // tensor_dim0_stride[31:0]
    g1[6] = 0;                                         // stride hi / dim1_stride lo
    g1[7] = 0;                                         // dim1_stride hi

    // ---- D# group 2 (128b), iterate mode
    v4i g2;
    g2[0] = 1;                                         // tensor_dim2 = 1
    g2[1] = 0;                                         // lds_addr_increment = 0 (reuse zeros)
    g2[2] = TILE_DWORDS;                               // global_addr_increment (elem units)
    g2[3] = (TDM_ITER - 1) << 16;                      // iterate_count in [127:112]

    // ---- D# group 3 (128b): unused dims -> 0
    v4i g3;
    g3[0] = 0; g3[1] = 0; g3[2] = 0; g3[3] = 0;

    // ---- extra descriptor operand required by the 6-arg toolchain builtin
    v8i g4;
    g4[0] = 0; g4[1] = 0; g4[2] = 0; g4[3] = 0;
    g4[4] = 0; g4[5] = 0; g4[6] = 0; g4[7] = 0;

    __builtin_amdgcn_tensor_store_from_lds(g0, g1, g2, g3, g4, /*cpol*/0);
    __builtin_amdgcn_s_wait_tensorcnt(0);
  }
}

// Fallback tail zeroing (not needed for the exact reference shape, kept for safety).
__global__ void ARNOLD_zero_tail(float* __restrict__ out, size_t start, size_t count) {
  size_t i = (size_t)blockIdx.x * blockDim.x + threadIdx.x;
  if (i < count) out[start + i] = 0.0f;
}

// One thread per (b, n) pair: compute bin, store 1.0f if bin < 100.
__global__ void ARNOLD_scatter_spikes(const float* __restrict__ trace,
                                      const float* __restrict__ center,
                                      float* __restrict__ out,
                                      int total /* B*N = 1048576 */) {
  int idx = blockIdx.x * blockDim.x + threadIdx.x;
  if (idx >= total) return;
  int b = idx >> 10;        // N = 1024
  int n = idx & 1023;
  int f = n >> 3;           // C = 8 centers
  int c = n & 7;

  float x   = trace[(b << 7) + f];   // F = 128
  float cen = center[c];

  float t = 2.5f * fabsf(x - cen);   // SCALING * |x - c|
  if (t > 10.0f) t = 10.1f;          // cutoff -> TIME_LENGTH + DT
  int bin = (int)(t / 0.1f + 1.0f);  // trunc == floor (t >= 0)

  if (bin < 100) {                   // TIME_STEPS
    out[(size_t)bin * (size_t)total + (size_t)idx] = 1.0f;
  }
}

extern "C" void kernel_launch(void* const* d_in, const int* in_sizes, int n_in,
                              void* d_out, int out_size, void* d_ws, size_t ws_size,
                              hipStream_t stream) {
  const float* trace  = (const float*)d_in[0];   // [1024, 128] f32
  const float* center = (const float*)d_in[1];   // [8] f32
  float* out = (float*)d_out;                    // [100, 1024, 1024]

  // ---- Pass 1: TDM zero-fill of the full output (256 KB per block).
  size_t bytes    = (size_t)out_size * sizeof(float);       // 419,430,400
  size_t nfull    = bytes / BLOCK_BYTES;                    // 1600 exact
  size_t rem_elts = (bytes - nfull * (size_t)BLOCK_BYTES) / sizeof(float);
  if (nfull > 0) {
    ARNOLD_zero_fill_tdm<<<(int)nfull, 256, 0, stream>>>(out);
  }
  if (rem_elts > 0) {
    size_t start = nfull * (BLOCK_BYTES / sizeof(float));
    int blks = (int)((rem_elts + 255) / 256);
    ARNOLD_zero_tail<<<blks, 256, 0, stream>>>(out, start, rem_elts);
  }

  // ---- Pass 2: scatter the one-hot ones (kernel boundary orders vs pass 1).
  int total = in_sizes[0] * ((n_in > 1) ? in_sizes[1] : 8); // 131072 * 8 = 1,048,576
  ARNOLD_scatter_spikes<<<(total + 255) / 256, 256, 0, stream>>>(trace, center, out, total);
}